// SRR_5334349382348
// MI455X (gfx1250) — compile-verified
//
#include <hip/hip_runtime.h>
#include <hip/hip_bf16.h>

// ---------------- problem constants (match reference) ----------------
#define N_SLICES 32
#define HS 160
#define WS_ 160
#define DD 160
#define HH 160
#define WW 160
#define KK 5
#define VOLN (DD * HH * WW)          // 4,096,000 (multiple of 128)
#define NSL (N_SLICES * HS * WS_)    // 819,200
#define NPART 256                    // stage-1 reduction blocks

typedef __attribute__((ext_vector_type(2))) float v2f;
typedef __attribute__((ext_vector_type(8))) float v8f;

// ---------------- helpers ----------------
__device__ __forceinline__ void atom_add_f32(float* p, float v) {
#if defined(__gfx1250__)
  unsafeAtomicAdd(p, v);   // lowers to global_atomic_add_f32 (L2-side fp32 atomic)
#else
  atomicAdd(p, v);
#endif
}

// Transformed sample point for slice n, pixel (h,w), psf tap k.
__device__ __forceinline__ void sample_point(const float* __restrict__ Tr,
                                             int h, int w, int k,
                                             float& qx, float& qy, float& qz) {
  const float px = (float)w - 79.5f;   // (WS-1)/2, S=1
  const float py = (float)h - 79.5f;
  const float pz = (float)k - 2.0f;    // (K-1)/2
  qx = Tr[0] * px + Tr[1] * py + Tr[2]  * pz + Tr[3]  + 79.5f;
  qy = Tr[4] * px + Tr[5] * py + Tr[6]  * pz + Tr[7]  + 79.5f;
  qz = Tr[8] * px + Tr[9] * py + Tr[10] * pz + Tr[11] + 79.5f;
}

// Trilinear corner indices + weights (weight forced to 0 when out of bounds,
// index clamped — matches the reference exactly).
__device__ __forceinline__ void corners(float qx, float qy, float qz,
                                        int idx[8], float wgt[8]) {
  const float x0f = floorf(qx), y0f = floorf(qy), z0f = floorf(qz);
  const float fx = qx - x0f, fy = qy - y0f, fz = qz - z0f;
  const int x0 = (int)x0f, y0 = (int)y0f, z0 = (int)z0f;
  const float wx[2] = {1.0f - fx, fx};
  const float wy[2] = {1.0f - fy, fy};
  const float wz[2] = {1.0f - fz, fz};
  int c = 0;
#pragma unroll
  for (int dz = 0; dz < 2; ++dz)
#pragma unroll
    for (int dy = 0; dy < 2; ++dy)
#pragma unroll
      for (int dx = 0; dx < 2; ++dx, ++c) {
        const int X = x0 + dx, Y = y0 + dy, Z = z0 + dz;
        const bool valid = ((unsigned)X < (unsigned)WW) &
                           ((unsigned)Y < (unsigned)HH) &
                           ((unsigned)Z < (unsigned)DD);
        const int Xc = min(max(X, 0), WW - 1);
        const int Yc = min(max(Y, 0), HH - 1);
        const int Zc = min(max(Z, 0), DD - 1);
        idx[c] = (Zc * HH + Yc) * WW + Xc;
        wgt[c] = valid ? (wx[dx] * wy[dy] * wz[dz]) : 0.0f;
      }
}

// ---------------- operator kernels ----------------
// A: y[n,h,w] = sum_k psf[k] * trilinear(vol, q(n,h,w,k))
__global__ void __launch_bounds__(256)
a_kernel(const float* __restrict__ T, const float* __restrict__ vol,
         const float* __restrict__ psf, float* __restrict__ y) {
  const int tid = blockIdx.x * 256 + threadIdx.x;
  if (tid >= NSL) return;
  const int n = tid / (HS * WS_);
  const int rem = tid - n * (HS * WS_);
  const int h = rem / WS_;
  const int w = rem - h * WS_;
  float Tr[12];
#pragma unroll
  for (int i = 0; i < 12; ++i) Tr[i] = T[n * 12 + i];
  float pk[KK];
#pragma unroll
  for (int k = 0; k < KK; ++k) pk[k] = psf[k];

  float out = 0.0f;
#pragma unroll
  for (int k = 0; k < KK; ++k) {
    float qx, qy, qz;
    sample_point(Tr, h, w, k, qx, qy, qz);
    int idx[8]; float wgt[8];
    corners(qx, qy, qz, idx, wgt);
    float acc = 0.0f;
#pragma unroll
    for (int c = 0; c < 8; ++c) acc += wgt[c] * vol[idx[c]];
    out += acc * pk[k];
  }
  y[tid] = out;
}

// At: vol += scatter of y[n,h,w]*psf[k] with trilinear weights (fp32 atomics).
__global__ void __launch_bounds__(256)
at_kernel(const float* __restrict__ T, const float* __restrict__ y,
          const float* __restrict__ psf, float* __restrict__ vol) {
  const int tid = blockIdx.x * 256 + threadIdx.x;
  if (tid >= NSL) return;
  const int n = tid / (HS * WS_);
  const int rem = tid - n * (HS * WS_);
  const int h = rem / WS_;
  const int w = rem - h * WS_;
  float Tr[12];
#pragma unroll
  for (int i = 0; i < 12; ++i) Tr[i] = T[n * 12 + i];
  const float yv = y[tid];
#pragma unroll
  for (int k = 0; k < KK; ++k) {
    const float contrib = yv * psf[k];
    float qx, qy, qz;
    sample_point(Tr, h, w, k, qx, qy, qz);
    int idx[8]; float wgt[8];
    corners(qx, qy, qz, idx, wgt);
#pragma unroll
    for (int c = 0; c < 8; ++c) {
      const float a = wgt[c] * contrib;
      if (a != 0.0f) atom_add_f32(vol + idx[c], a);
    }
  }
}

// ---------------- vectorized (float4 / B128) elementwise kernels ----------------
// All CG vectors are L2-resident; float4 quarters the instruction count per byte.
__global__ void __launch_bounds__(256) zero4_kernel(float4* p, int n4) {
  int i = blockIdx.x * 256 + threadIdx.x;
  if (i < n4) p[i] = make_float4(0.f, 0.f, 0.f, 0.f);
}
__global__ void __launch_bounds__(256) copy4_kernel(float4* d, const float4* s, int n4) {
  int i = blockIdx.x * 256 + threadIdx.x;
  if (i < n4) d[i] = s[i];
}
// r = b - Av ; p = r
__global__ void __launch_bounds__(256)
residual4_kernel(float4* r, float4* p, const float4* b, const float4* Av, int n4) {
  int i = blockIdx.x * 256 + threadIdx.x;
  if (i < n4) {
    float4 bb = b[i], aa = Av[i], v;
    v.x = bb.x - aa.x; v.y = bb.y - aa.y; v.z = bb.z - aa.z; v.w = bb.w - aa.w;
    r[i] = v; p[i] = v;
  }
}
// alpha = s[iRR]/s[iPAP]; x += alpha p; r -= alpha Ap
__global__ void __launch_bounds__(256)
update_xr4(float4* x, float4* r, const float4* p, const float4* Ap,
           const float* __restrict__ s, int iRR, int iPAP, int n4) {
  const float alpha = s[iRR] / s[iPAP];
  int i = blockIdx.x * 256 + threadIdx.x;
  if (i < n4) {
    float4 xv = x[i], pv = p[i], rv = r[i], av = Ap[i];
    xv.x += alpha * pv.x; xv.y += alpha * pv.y; xv.z += alpha * pv.z; xv.w += alpha * pv.w;
    rv.x -= alpha * av.x; rv.y -= alpha * av.y; rv.z -= alpha * av.z; rv.w -= alpha * av.w;
    x[i] = xv; r[i] = rv;
  }
}
// beta = s[iNew]/s[iOld]; p = r + beta p
__global__ void __launch_bounds__(256)
update_p4(float4* p, const float4* r, const float* __restrict__ s,
          int iNew, int iOld, int n4) {
  const float beta = s[iNew] / s[iOld];
  int i = blockIdx.x * 256 + threadIdx.x;
  if (i < n4) {
    float4 rv = r[i], pv = p[i];
    pv.x = rv.x + beta * pv.x; pv.y = rv.y + beta * pv.y;
    pv.z = rv.z + beta * pv.z; pv.w = rv.w + beta * pv.w;
    p[i] = pv;
  }
}
__global__ void __launch_bounds__(256) relu4_kernel(float4* x, int n4) {
  int i = blockIdx.x * 256 + threadIdx.x;
  if (i < n4) {
    float4 v = x[i];
    v.x = fmaxf(v.x, 0.f); v.y = fmaxf(v.y, 0.f);
    v.z = fmaxf(v.z, 0.f); v.w = fmaxf(v.w, 0.f);
    x[i] = v;
  }
}

// ---------------- WMMA-backed dot-product reduction ----------------
// D = ones(16x4) x B(4x16) + C : every row of D is the running column-sum of
// all B data fed in; 64 products absorbed per WMMA. Two independent
// accumulators hide the WMMA RAW latency (128 elements/iter/wave).
// Sum of accumulator VGPR0 over 32 lanes = 2x the wave total (rows 0 and 8).
__global__ void __launch_bounds__(256)
dot_stage1(const float* __restrict__ a, const float* __restrict__ b,
           float* __restrict__ partials, int n) {  // n must be a multiple of 64
  const int lane = threadIdx.x & 31;
  const int waveInBlk = threadIdx.x >> 5;
  const int wavesPerBlk = blockDim.x >> 5;
  const int gwave = blockIdx.x * wavesPerBlk + waveInBlk;
  const int gstride = gridDim.x * wavesPerBlk;

  v8f c0 = {}, c1 = {};
  v2f ones; ones[0] = 1.0f; ones[1] = 1.0f;

  int base = gwave * 128;
  for (; base + 128 <= n; base += gstride * 128) {
    const int i0 = base + lane * 2;
    const int i1 = base + 64 + lane * 2;
    v2f p0, p1;
    p0[0] = a[i0] * b[i0];
    p0[1] = a[i0 + 1] * b[i0 + 1];
    p1[0] = a[i1] * b[i1];
    p1[1] = a[i1 + 1] * b[i1 + 1];
    c0 = __builtin_amdgcn_wmma_f32_16x16x4_f32(false, ones, false, p0,
                                               (short)0, c0, false, false);
    c1 = __builtin_amdgcn_wmma_f32_16x16x4_f32(false, ones, false, p1,
                                               (short)0, c1, false, false);
  }
  if (base + 64 <= n) {  // leftover 64-chunk (wave-uniform condition)
    const int i0 = base + lane * 2;
    v2f p0;
    p0[0] = a[i0] * b[i0];
    p0[1] = a[i0 + 1] * b[i0 + 1];
    c0 = __builtin_amdgcn_wmma_f32_16x16x4_f32(false, ones, false, p0,
                                               (short)0, c0, false, false);
  }

  float s = c0[0] + c1[0];
#pragma unroll
  for (int off = 16; off >= 1; off >>= 1) s += __shfl_xor(s, off, 32);
  s *= 0.5f;  // rows 0 and 8 both counted

  __shared__ float sw[8];
  if (lane == 0) sw[waveInBlk] = s;
  __syncthreads();
  if (threadIdx.x == 0) {
    float t = 0.0f;
    for (int i = 0; i < wavesPerBlk; ++i) t += sw[i];
    partials[blockIdx.x] = t;
  }
}

// Stage 2: stage the partials into LDS with the gfx1250 async-to-LDS path
// (ASYNCcnt-tracked), then tree-reduce.
__global__ void __launch_bounds__(256)
dot_stage2(const float* __restrict__ partials, int n, float* __restrict__ out) {
  __shared__ float sm[256];
#if defined(__gfx1250__) &&                                         \
    __has_builtin(__builtin_amdgcn_global_load_async_to_lds_b32) && \
    __has_builtin(__builtin_amdgcn_s_wait_asynccnt)
  // Builtin signature (from hipcc diagnostic): (global int*, LDS int*, imm, imm)
  typedef __attribute__((address_space(1))) int as1_int;
  typedef __attribute__((address_space(3))) int as3_int;
  if ((int)threadIdx.x < n) {
    __builtin_amdgcn_global_load_async_to_lds_b32(
        (as1_int*)(partials + threadIdx.x), (as3_int*)(&sm[threadIdx.x]), 0, 0);
  } else {
    sm[threadIdx.x] = 0.0f;
  }
  __builtin_amdgcn_s_wait_asynccnt(0);
  __syncthreads();
#else
  sm[threadIdx.x] = ((int)threadIdx.x < n) ? partials[threadIdx.x] : 0.0f;
  __syncthreads();
#endif
  for (int off = 128; off >= 1; off >>= 1) {
    if (threadIdx.x < (unsigned)off) sm[threadIdx.x] += sm[threadIdx.x + off];
    __syncthreads();
  }
  if (threadIdx.x == 0) out[0] = sm[0];
}

// ---------------- orchestration ----------------
extern "C" void kernel_launch(void* const* d_in, const int* in_sizes, int n_in,
                              void* d_out, int out_size, void* d_ws, size_t ws_size,
                              hipStream_t stream) {
  const float* transforms = (const float*)d_in[0];  // (32,3,4)
  const float* slices     = (const float*)d_in[1];  // (32,160,160)
  const float* volume     = (const float*)d_in[2];  // (160,160,160)
  const float* psf        = (const float*)d_in[3];  // (5,)

  float* x = (float*)d_out;            // CG iterate lives in the output buffer
  float* b        = (float*)d_ws;      // workspace layout (fp32):
  float* r        = b  + VOLN;
  float* p        = r  + VOLN;
  float* Ap       = p  + VOLN;
  float* img      = Ap + VOLN;         // A(...) output, (32,160,160)
  float* partials = img + NSL;
  float* scal     = partials + NPART;  // [rr0, pAp1, rr1, pAp2, rr2, pAp3]

  const dim3 blk(256);
  const int NV4  = VOLN / 4;            // 1,024,000
  const int g4   = (NV4 + 255) / 256;   // 4000 blocks
  const int gImg = (NSL + 255) / 256;   // 3200 blocks

  auto DOT = [&](const float* aa, const float* bb, float* slot) {
    dot_stage1<<<NPART, blk, 0, stream>>>(aa, bb, partials, VOLN);
    dot_stage2<<<1, blk, 0, stream>>>(partials, NPART, slot);
  };
  auto ATA = [&](const float* v, float* outVol) {  // outVol = At(A(v))
    a_kernel<<<gImg, blk, 0, stream>>>(transforms, v, psf, img);
    zero4_kernel<<<g4, blk, 0, stream>>>((float4*)outVol, NV4);
    at_kernel<<<gImg, blk, 0, stream>>>(transforms, img, psf, outVol);
  };

  // b = At(slices)
  zero4_kernel<<<g4, blk, 0, stream>>>((float4*)b, NV4);
  at_kernel<<<gImg, blk, 0, stream>>>(transforms, slices, psf, b);

  // x = volume ; r = p = b - AtA(x) ; rr0 = <r,r>
  copy4_kernel<<<g4, blk, 0, stream>>>((float4*)x, (const float4*)volume, NV4);
  ATA(x, Ap);
  residual4_kernel<<<g4, blk, 0, stream>>>((float4*)r, (float4*)p,
                                           (const float4*)b, (const float4*)Ap, NV4);
  DOT(r, r, scal + 0);

  // --- CG iteration 1 ---
  ATA(p, Ap);
  DOT(p, Ap, scal + 1);                                        // pAp1
  update_xr4<<<g4, blk, 0, stream>>>((float4*)x, (float4*)r, (const float4*)p,
                                     (const float4*)Ap, scal, 0, 1, NV4);
  DOT(r, r, scal + 2);                                         // rr1
  update_p4<<<g4, blk, 0, stream>>>((float4*)p, (const float4*)r, scal, 2, 0, NV4);

  // --- CG iteration 2 ---
  ATA(p, Ap);
  DOT(p, Ap, scal + 3);                                        // pAp2
  update_xr4<<<g4, blk, 0, stream>>>((float4*)x, (float4*)r, (const float4*)p,
                                     (const float4*)Ap, scal, 2, 3, NV4);
  DOT(r, r, scal + 4);                                         // rr2
  update_p4<<<g4, blk, 0, stream>>>((float4*)p, (const float4*)r, scal, 4, 2, NV4);

  // --- CG iteration 3 (only x-update affects the result) ---
  ATA(p, Ap);
  DOT(p, Ap, scal + 5);                                        // pAp3
  update_xr4<<<g4, blk, 0, stream>>>((float4*)x, (float4*)r, (const float4*)p,
                                     (const float4*)Ap, scal, 4, 5, NV4);

  relu4_kernel<<<g4, blk, 0, stream>>>((float4*)x, NV4);
}